// GaussianKernel_7327214207092
// MI455X (gfx1250) — compile-verified
//
#include <hip/hip_runtime.h>
#include <hip/hip_bf16.h>

typedef __attribute__((ext_vector_type(2))) float v2f;
typedef __attribute__((ext_vector_type(8))) float v8f;

#define KDIM 64   // feature dimension (fixed by reference)

// ---------------------------------------------------------------------------
// Kernel 1: row squared-norms  n[i] = sum_k H[i][k]^2   (K = 64)
// ---------------------------------------------------------------------------
__global__ void row_norms_kernel(const float* __restrict__ H,
                                 float* __restrict__ norms, int n) {
    int i = blockIdx.x * blockDim.x + threadIdx.x;
    if (i >= n) return;
    const float4* p = (const float4*)(H + (size_t)i * KDIM);
    float s = 0.f;
#pragma unroll
    for (int j = 0; j < KDIM / 4; ++j) {
        float4 v = p[j];
        s += v.x * v.x + v.y * v.y + v.z * v.z + v.w * v.w;
    }
    norms[i] = s;
}

// ---------------------------------------------------------------------------
// Kernel 2: one wave computes a 16x64 strip of the output using
// V_WMMA_F32_16X16X4_F32 (fp32 matrix pipe, full precision).
//
// A (16x4 fp32) per-lane layout: lane&15 = M row, lane>>4 selects K pair:
//   v0 = H1[M, 4*kk + 2*half], v1 = H1[M, 4*kk + 2*half + 1]
// B (4x16 fp32) = H2^T tile, same addressing with N = lane&15 as the H2 row.
// C/D (16x16 fp32, 8 VGPRs): VGPR r, lane half -> M = r + 8*half, N = lane&15.
// ---------------------------------------------------------------------------
__global__ void __launch_bounds__(32)
gauss_wmma_kernel(const float* __restrict__ H1, const float* __restrict__ H2,
                  const float* __restrict__ n1, const float* __restrict__ n2,
                  float* __restrict__ out, int N2) {
    const int lane = threadIdx.x;        // 0..31, all active (WMMA needs EXEC all-1)
    const int half = lane >> 4;          // 0 or 1
    const int lrow = lane & 15;

    const int tm = blockIdx.x;           // 16-row tile index
    const int cn = blockIdx.y * 64;      // 64-column strip base

    // ---- load all A fragments for this row tile (row reused 4x) ----
    const float* pa = H1 + (size_t)(tm * 16 + lrow) * KDIM + 2 * half;
    v2f a[KDIM / 4];
#pragma unroll
    for (int kk = 0; kk < KDIM / 4; ++kk)
        a[kk] = *(const v2f*)(pa + 4 * kk);

    // ---- B fragment base pointers for the 4 column tiles ----
    const float* pb[4];
#pragma unroll
    for (int t = 0; t < 4; ++t)
        pb[t] = H2 + (size_t)(cn + t * 16 + lrow) * KDIM + 2 * half;

    // ---- 4 independent accumulator chains, k innermost-interleaved so the
    //      XDL pipe always has an independent WMMA to issue ----
    v8f acc[4];
#pragma unroll
    for (int t = 0; t < 4; ++t) acc[t] = (v8f){0.f,0.f,0.f,0.f,0.f,0.f,0.f,0.f};

#pragma unroll
    for (int kk = 0; kk < KDIM / 4; ++kk) {
#pragma unroll
        for (int t = 0; t < 4; ++t) {
            v2f b = *(const v2f*)(pb[t] + 4 * kk);
            acc[t] = __builtin_amdgcn_wmma_f32_16x16x4_f32(
                /*neg_a=*/false, a[kk], /*neg_b=*/false, b,
                /*c_mod=*/(short)0, acc[t],
                /*reuse_a=*/false, /*reuse_b=*/false);
        }
    }

    // ---- epilogue: S = exp(2*dot - |xi|^2 - |xj|^2), non-temporal stores ----
    // rows covered by this lane: tm*16 + 8*half + r, r = 0..7 (consecutive)
    float rn[8];
    const float* pn1 = n1 + tm * 16 + 8 * half;
#pragma unroll
    for (int r = 0; r < 8; ++r) rn[r] = pn1[r];

#pragma unroll
    for (int t = 0; t < 4; ++t) {
        const int col = cn + t * 16 + lrow;
        const float cnorm = n2[col];
#pragma unroll
        for (int r = 0; r < 8; ++r) {
            float arg = 2.0f * acc[t][r] - rn[r] - cnorm;  // = -D2 (KERNEL_PARA=1)
            float val = __expf(arg);
            size_t idx = (size_t)(tm * 16 + 8 * half + r) * (size_t)N2 + (size_t)col;
            __builtin_nontemporal_store(val, out + idx);
        }
    }
}

// ---------------------------------------------------------------------------
extern "C" void kernel_launch(void* const* d_in, const int* in_sizes, int n_in,
                              void* d_out, int out_size, void* d_ws, size_t ws_size,
                              hipStream_t stream) {
    const float* H1 = (const float*)d_in[0];
    const float* H2 = (const float*)d_in[1];
    float* out = (float*)d_out;

    const int N1 = in_sizes[0] / KDIM;   // 8192
    const int N2 = in_sizes[1] / KDIM;   // 8192

    float* norms1 = (float*)d_ws;
    float* norms2 = norms1 + N1;

    row_norms_kernel<<<(N1 + 255) / 256, 256, 0, stream>>>(H1, norms1, N1);
    row_norms_kernel<<<(N2 + 255) / 256, 256, 0, stream>>>(H2, norms2, N2);

    dim3 grid(N1 / 16, N2 / 64);         // (512, 128) single-wave workgroups
    gauss_wmma_kernel<<<grid, 32, 0, stream>>>(H1, H2, norms1, norms2, out, N2);
}